// HingeLoss_23837068493403
// MI455X (gfx1250) — compile-verified
//
#include <hip/hip_runtime.h>
#include <hip/hip_bf16.h>

typedef __attribute__((ext_vector_type(2))) float v2f;
typedef __attribute__((ext_vector_type(8))) float v8f;

#define WAVES_PER_BLOCK 2
#define TILE 16
#define ROWPAD 132   // 128 floats + 4-float pad: row stride 528B -> 4-bank rotation, float4-aligned

// order-preserving float<->uint encoding for atomicMax on signed floats
__device__ __forceinline__ unsigned ordEnc(float f) {
    unsigned b = __float_as_uint(f);
    return (b & 0x80000000u) ? ~b : (b | 0x80000000u);
}
__device__ __forceinline__ float ordDec(unsigned u) {
    return (u & 0x80000000u) ? __uint_as_float(u ^ 0x80000000u)
                             : __uint_as_float(~u);
}

__global__ void init_ws(float* p_sum, float* p_cnt, unsigned* n_maxb, unsigned* n_cnt,
                        float* out, int N) {
    int i = blockIdx.x * blockDim.x + threadIdx.x;
    if (i == 0) out[0] = 0.0f;
    for (; i < N; i += gridDim.x * blockDim.x) {
        p_sum[i] = 0.0f;
        p_cnt[i] = 0.0f;
        n_maxb[i] = 0u;   // encodes "below every real encoded value" (acts as -inf sentinel)
        n_cnt[i] = 0u;
    }
}

// MODE 0: positive edges (segment sum + count). MODE 1: negative edges (segment max + count).
template <int MODE>
__global__ __launch_bounds__(WAVES_PER_BLOCK * 32)
void edge_dot_kernel(const float* __restrict__ emb, const int* __restrict__ inv_idx,
                     const int* __restrict__ src_idx, const int* __restrict__ dst_idx,
                     int E,
                     float* __restrict__ seg_sum, float* __restrict__ seg_cntf,
                     unsigned* __restrict__ seg_maxb, unsigned* __restrict__ seg_cntu)
{
    __shared__ float lds[WAVES_PER_BLOCK * 2 * TILE * ROWPAD];
    const int lane = threadIdx.x & 31;
    const int wave = threadIdx.x >> 5;
    float* ldsA = lds + wave * (2 * TILE * ROWPAD);
    float* ldsB = ldsA + TILE * ROWPAD;

    const int numTiles  = (E + TILE - 1) / TILE;
    const int waveGlob  = blockIdx.x * WAVES_PER_BLOCK + wave;
    const int waveStride = gridDim.x * WAVES_PER_BLOCK;

    for (int t = waveGlob; t < numTiles; t += waveStride) {
        const int base = t * TILE;

        // lanes 0..15 fetch the 16 edges of this tile (clamped; dupes skipped at atomic stage)
        int e = base + (lane & 15);
        if (e >= E) e = E - 1;
        const int eSeg = src_idx[e];           // segment id = raw src index
        const int eDst = dst_idx[e];
        const int rowS = inv_idx[eSeg];
        const int rowD = inv_idx[eDst];

        // speculative prefetch of the next tile's index data (global_prefetch_b8)
        __builtin_prefetch(src_idx + base + waveStride * TILE, 0, 0);
        __builtin_prefetch(dst_idx + base + waveStride * TILE, 0, 0);

        // cooperative gather: 16 src rows + 16 dst rows (512B each) into LDS via b128 ops
        #pragma unroll 4
        for (int r = 0; r < TILE; ++r) {
            const int rs = __shfl(rowS, r, 32);
            const int rd = __shfl(rowD, r, 32);
            const float4 a4 = ((const float4*)(emb + (size_t)rs * 128))[lane];
            const float4 b4 = ((const float4*)(emb + (size_t)rd * 128))[lane];
            *(float4*)(ldsA + r * ROWPAD + lane * 4) = a4;
            *(float4*)(ldsB + r * ROWPAD + lane * 4) = b4;
        }
        __builtin_amdgcn_wave_barrier();
        asm volatile("s_wait_dscnt 0x0" ::: "memory");   // per-wave LDS producer/consumer fence

        // D(16x16) = A(16x128, src rows) x B(128x16, dst rows K-major), K in steps of 4.
        // A frag (ISA 7.12.2): lane -> M = lane&15; VGPR0/1 hold K = khalf, khalf+1.
        // B frag:              lane -> N = lane&15; VGPR0/1 hold K = khalf, khalf+1.
        v8f c = {};
        const int mrow  = lane & 15;
        const int khalf = (lane >> 4) << 1;   // 0 for lanes 0-15, 2 for lanes 16-31
        #pragma unroll
        for (int kk = 0; kk < 128; kk += 4) {
            v2f a, b;
            a.x = ldsA[mrow * ROWPAD + kk + khalf + 0];
            a.y = ldsA[mrow * ROWPAD + kk + khalf + 1];
            b.x = ldsB[mrow * ROWPAD + kk + khalf + 0];
            b.y = ldsB[mrow * ROWPAD + kk + khalf + 1];
            c = __builtin_amdgcn_wmma_f32_16x16x4_f32(false, a, false, b,
                                                      (short)0, c, false, false);
        }

        // Diagonal extraction: D[m][m]. C/D layout: VGPR v, lanes 0-15 -> (M=v, N=lane),
        // lanes 16-31 -> (M=v+8, N=lane-16). So m=0..7 lives in lane m at c[m];
        // m=8..15 lives in lane m+16 at c[m-8].
        const bool ownsDiag = (lane < 8) || (lane >= 24);
        const int  vsel = (lane < 8) ? lane : (lane - 24);
        const int  eoff = (lane < 8) ? lane : (lane - 16);
        float d = c[0];
        #pragma unroll
        for (int v = 1; v < 8; ++v) d = (vsel == v) ? c[v] : d;

        const int seg  = __shfl(eSeg, eoff, 32);
        const int edge = base + eoff;
        if (ownsDiag && edge < E) {
            if (MODE == 0) {
                atomicAdd(&seg_sum[seg], d);
                atomicAdd(&seg_cntf[seg], 1.0f);
            } else {
                atomicMax(&seg_maxb[seg], ordEnc(d));
                atomicAdd(&seg_cntu[seg], 1u);
            }
        }
        asm volatile("s_wait_dscnt 0x0" ::: "memory");   // safe LDS reuse next iteration
    }
}

__global__ void finalize_kernel(const float* __restrict__ p_sum, const float* __restrict__ p_cnt,
                                const unsigned* __restrict__ n_maxb, const unsigned* __restrict__ n_cnt,
                                float* __restrict__ out, int N)
{
    __shared__ float red[256];
    float acc = 0.0f;
    for (int i = blockIdx.x * blockDim.x + threadIdx.x; i < N; i += gridDim.x * blockDim.x) {
        const float pd = p_sum[i] / fmaxf(p_cnt[i], 1.0f);
        const float nd = (n_cnt[i] > 0u) ? ordDec(n_maxb[i]) : 0.0f;
        acc += fmaxf(nd - pd + 1.0f, 0.0f);
    }
    red[threadIdx.x] = acc;
    __syncthreads();
    for (int s = blockDim.x / 2; s > 0; s >>= 1) {
        if ((int)threadIdx.x < s) red[threadIdx.x] += red[threadIdx.x + s];
        __syncthreads();
    }
    if (threadIdx.x == 0) atomicAdd(out, red[0] / (float)N);
}

extern "C" void kernel_launch(void* const* d_in, const int* in_sizes, int n_in,
                              void* d_out, int out_size, void* d_ws, size_t ws_size,
                              hipStream_t stream) {
    const float* emb     = (const float*)d_in[0];
    const int*   inv_idx = (const int*)d_in[1];
    const int*   pos_src = (const int*)d_in[2];
    const int*   pos_dst = (const int*)d_in[3];
    const int*   neg_src = (const int*)d_in[4];
    const int*   neg_dst = (const int*)d_in[5];
    const int N = in_sizes[0] / 128;   // emb is (N, 128)
    const int E = in_sizes[2];
    float* out = (float*)d_out;

    // workspace carve: p_sum[N] f32 | p_cnt[N] f32 | n_maxb[N] u32 | n_cnt[N] u32
    float*    p_sum  = (float*)d_ws;
    float*    p_cnt  = p_sum + N;
    unsigned* n_maxb = (unsigned*)(p_cnt + N);
    unsigned* n_cnt  = n_maxb + N;

    init_ws<<<512, 256, 0, stream>>>(p_sum, p_cnt, n_maxb, n_cnt, out, N);

    const int numTiles = (E + TILE - 1) / TILE;
    int blocks = (numTiles + WAVES_PER_BLOCK - 1) / WAVES_PER_BLOCK;
    if (blocks > 2048) blocks = 2048;

    edge_dot_kernel<0><<<blocks, WAVES_PER_BLOCK * 32, 0, stream>>>(
        emb, inv_idx, pos_src, pos_dst, E, p_sum, p_cnt, n_maxb, n_cnt);
    edge_dot_kernel<1><<<blocks, WAVES_PER_BLOCK * 32, 0, stream>>>(
        emb, inv_idx, neg_src, neg_dst, E, p_sum, p_cnt, n_maxb, n_cnt);

    finalize_kernel<<<256, 256, 0, stream>>>(p_sum, p_cnt, n_maxb, n_cnt, out, N);
}